// HyperbolicMapper_62809601736877
// MI455X (gfx1250) — compile-verified
//
#include <hip/hip_runtime.h>
#include <hip/hip_bf16.h>
#include <math.h>

// Problem constants from the reference
#define DN 4096
#define KK 16
#define LN_EPS 1e-5f
#define SN_EPS 1e-12f
#define MIN_NORM 1e-15f

typedef float v2f __attribute__((ext_vector_type(2)));
typedef float v4f __attribute__((ext_vector_type(4)));
typedef float v8f __attribute__((ext_vector_type(8)));

// Workspace layout (floats):
//  ws[0]                : fused scalar s = sigmoid(log_scale)*0.5 / sigma
//  ws[16..31]           : a[k] = sum_d gamma[d]*W[k,d]
//  ws[32..47]           : b[k] = sum_d beta[d]*W[k,d]
//  ws[64 .. 64+65536)   : packB — WMMA B fragments of (gamma ⊙ W)^T, packed
//                         pair-of-steps-major so each lane loads 2 steps with
//                         one b128:
//                         pair p in [0,512), lane l in [0,32), word w in [0,4):
//                           step = 2p + (w>>1), j = w&1
//                           d = 4*step + 2*(l>>4) + j, k = l&15
//                           packB[(p*32+l)*4+w] = gamma[d]*W[k,d]
//                         (B 4x16 layout: VGPR0 = rows K=0 / K=2 per half-wave,
//                          VGPR1 = rows K=1 / K=3 — mirrors the documented A/C striping)

__global__ __launch_bounds__(256)
void hm_prep(const float* __restrict__ gamma, const float* __restrict__ beta,
             const float* __restrict__ W, const float* __restrict__ u,
             const float* __restrict__ log_scale, float* __restrict__ ws)
{
    __shared__ float t[DN];       // W^T u (unnormalized), 16 KB
    __shared__ float red;         // ||t||^2
    __shared__ float acc[48];     // [0:16)=Wv, [16:32)=a[k], [32:48)=b[k]

    const int tid = threadIdx.x;
    if (tid < 48) acc[tid] = 0.0f;
    if (tid == 0) red = 0.0f;
    __syncthreads();

    // ---- phase 1: t[d] = sum_k W[k,d]*u[k], accumulate ||t||^2 ----
    float uk[KK];
    #pragma unroll
    for (int k = 0; k < KK; ++k) uk[k] = u[k];

    float ssq = 0.0f;
    for (int d = tid; d < DN; d += 256) {
        float td = 0.0f;
        #pragma unroll
        for (int k = 0; k < KK; ++k) td = fmaf(W[k * DN + d], uk[k], td);
        t[d] = td;
        ssq = fmaf(td, td, ssq);
    }
    #pragma unroll
    for (int m = 16; m >= 1; m >>= 1) ssq += __shfl_xor(ssq, m, 32);
    if ((tid & 31) == 0) atomicAdd(&red, ssq);
    __syncthreads();

    const float inv = 1.0f / (sqrtf(red) + SN_EPS);   // v = t * inv

    // ---- phase 2: Wv[k], a[k], b[k] ----
    {
        float p[KK], ga[KK], bb[KK];
        #pragma unroll
        for (int k = 0; k < KK; ++k) { p[k] = 0.0f; ga[k] = 0.0f; bb[k] = 0.0f; }
        for (int d = tid; d < DN; d += 256) {
            const float tv = t[d] * inv;
            const float g = gamma[d];
            const float be = beta[d];
            #pragma unroll
            for (int k = 0; k < KK; ++k) {
                const float w = W[k * DN + d];
                p[k]  = fmaf(w, tv, p[k]);
                ga[k] = fmaf(w, g,  ga[k]);
                bb[k] = fmaf(w, be, bb[k]);
            }
        }
        #pragma unroll
        for (int k = 0; k < KK; ++k) {
            atomicAdd(&acc[k],      p[k]);
            atomicAdd(&acc[16 + k], ga[k]);
            atomicAdd(&acc[32 + k], bb[k]);
        }
    }
    __syncthreads();

    if (tid == 0) {
        float nv2 = 0.0f;
        #pragma unroll
        for (int k = 0; k < KK; ++k) nv2 = fmaf(acc[k], acc[k], nv2);
        const float nv = sqrtf(nv2);
        // u_new = Wv/(||Wv||+eps);  sigma = u_new . Wv = ||Wv||^2/(||Wv||+eps)
        const float sigma = nv2 / (nv + SN_EPS);
        const float ls = log_scale[0];
        const float sig = 1.0f / (1.0f + expf(-ls));
        ws[0] = (sig * 0.5f) / sigma;
    }
    if (tid < 16) {
        ws[16 + tid] = acc[16 + tid];
        ws[32 + tid] = acc[32 + tid];
    }

    // ---- phase 3: pack B fragments (pair-of-steps-major, b128 per lane) ----
    float* __restrict__ pb = ws + 64;
    for (int idx = tid; idx < (DN / 4) * 32 * 2; idx += 256) {
        const int w    = idx & 3;
        const int lane = (idx >> 2) & 31;
        const int pair = idx >> 7;
        const int step = pair * 2 + (w >> 1);
        const int j    = w & 1;
        const int d    = step * 4 + ((lane >> 4) << 1) + j;
        const int k    = lane & 15;
        pb[idx] = gamma[d] * W[k * DN + d];
    }
}

// Main fused kernel: one wave per 16-row tile. Streams x once (non-temporal),
// accumulates sum / sumsq as packed v2f on VALU while v_wmma_f32_16x16x4_f32
// accumulates P = x * (gamma ⊙ W)^T on the matrix pipe. Epilogue: LN finalize,
// spectral scale, Poincare expmap0, store [16 x 16] tile.
__global__ __launch_bounds__(256)
void hm_main(const float* __restrict__ x, const float* __restrict__ ws,
             float* __restrict__ out)
{
    const int lane = threadIdx.x & 31;
    const int wave = threadIdx.x >> 5;
    const int tile = blockIdx.x * 8 + wave;
    const int rowBase = tile * 16;
    const int half = lane >> 4;           // 0: K=0,1 of A ; 1: K=2,3
    const int myRow = rowBase + (lane & 15);

    const float* __restrict__ xp  = x + (size_t)myRow * DN + half * 2;
    const float* __restrict__ pbp = ws + 64 + lane * 4;

    v8f c0 = {}; v8f c1 = {}; v8f c2 = {}; v8f c3 = {};
    v2f vs0 = {0.0f, 0.0f}, vs1 = {0.0f, 0.0f};   // packed sum accumulators
    v2f vq0 = {0.0f, 0.0f}, vq1 = {0.0f, 0.0f};   // packed sumsq accumulators

    #pragma unroll 2
    for (int p = 0; p < DN / 8; p += 2) {          // each p covers 2 k-steps
        // A fragments: x[row, 4*step + 2*half + {0,1}] (ISA 16x4 f32 A layout)
        v2f a0 = __builtin_nontemporal_load((const v2f*)(xp + 8 * p));
        v2f a1 = __builtin_nontemporal_load((const v2f*)(xp + 8 * p + 4));
        v2f a2 = __builtin_nontemporal_load((const v2f*)(xp + 8 * p + 8));
        v2f a3 = __builtin_nontemporal_load((const v2f*)(xp + 8 * p + 12));
        // B fragments: one b128 per lane covers 2 steps (L2-resident stream)
        v4f B0 = *(const v4f*)(pbp + 128 * p);
        v4f B1 = *(const v4f*)(pbp + 128 * p + 128);
        v2f b0 = __builtin_shufflevector(B0, B0, 0, 1);
        v2f b1 = __builtin_shufflevector(B0, B0, 2, 3);
        v2f b2 = __builtin_shufflevector(B1, B1, 0, 1);
        v2f b3 = __builtin_shufflevector(B1, B1, 2, 3);

        c0 = __builtin_amdgcn_wmma_f32_16x16x4_f32(false, a0, false, b0, (short)0, c0, false, false);
        c1 = __builtin_amdgcn_wmma_f32_16x16x4_f32(false, a1, false, b1, (short)0, c1, false, false);
        c2 = __builtin_amdgcn_wmma_f32_16x16x4_f32(false, a2, false, b2, (short)0, c2, false, false);
        c3 = __builtin_amdgcn_wmma_f32_16x16x4_f32(false, a3, false, b3, (short)0, c3, false, false);

        vs0 += a0; vs1 += a1; vs0 += a2; vs1 += a3;        // v_pk_add_f32
        vq0 += a0 * a0; vq1 += a1 * a1;                    // v_pk_fma_f32
        vq0 += a2 * a2; vq1 += a3 * a3;
    }

    v8f c = (c0 + c1) + (c2 + c3);        // P[row, k] in C layout
    float sum = (vs0.x + vs0.y) + (vs1.x + vs1.y);
    float ssq = (vq0.x + vq0.y) + (vq1.x + vq1.y);

    // Merge half-row stats: lanes L and L^16 together cover row (L&15)
    sum += __shfl_xor(sum, 16, 32);
    ssq += __shfl_xor(ssq, 16, 32);
    const float mean = sum * (1.0f / (float)DN);
    const float var  = ssq * (1.0f / (float)DN) - mean * mean;
    const float rstd = rsqrtf(var + LN_EPS);

    const float s  = ws[0];
    const float ak = ws[16 + (lane & 15)];
    const float bk = ws[32 + (lane & 15)];

    // C layout: element j of lane L is row m = j + 8*half, col k = lane&15
    float y[8];
    #pragma unroll
    for (int j = 0; j < 8; ++j) {
        const int src = j + half * 8;                 // lane holding row m's stats
        const float m = __shfl(mean, src, 32);
        const float r = __shfl(rstd, src, 32);
        y[j] = (r * (c[j] - m * ak) + bk) * s;
    }

    // Poincare expmap0: per-row L2 norm over the 16 k-values (one half-wave)
    #pragma unroll
    for (int j = 0; j < 8; ++j) {
        float q = y[j] * y[j];
        q += __shfl_xor(q, 1, 32);
        q += __shfl_xor(q, 2, 32);
        q += __shfl_xor(q, 4, 32);
        q += __shfl_xor(q, 8, 32);
        const float vn = fmaxf(sqrtf(q), MIN_NORM);
        const float f = tanhf(vn) / vn;
        out[(size_t)(rowBase + j + 8 * half) * KK + (lane & 15)] = y[j] * f;
    }
}

extern "C" void kernel_launch(void* const* d_in, const int* in_sizes, int n_in,
                              void* d_out, int out_size, void* d_ws, size_t ws_size,
                              hipStream_t stream) {
    const float* x         = (const float*)d_in[0];
    const float* gamma     = (const float*)d_in[1];
    const float* beta      = (const float*)d_in[2];
    const float* W         = (const float*)d_in[3];
    const float* u         = (const float*)d_in[4];
    const float* log_scale = (const float*)d_in[5];
    float* out = (float*)d_out;
    float* ws  = (float*)d_ws;

    hm_prep<<<1, 256, 0, stream>>>(gamma, beta, W, u, log_scale, ws);
    // 65536 rows / 16 rows-per-wave / 8 waves-per-block = 512 blocks
    hm_main<<<512, 256, 0, stream>>>(x, ws, out);
}